// VerticalSelfAttention_83966610637159
// MI455X (gfx1250) — compile-verified
//
#include <hip/hip_runtime.h>
#include <math.h>

#define E_  256
#define RR_ 255

typedef float v2f __attribute__((ext_vector_type(2)));
typedef float v8f __attribute__((ext_vector_type(8)));

// workspace layout (float offsets)
#define OFF_R   0u                       // R[h][e] : 4*256
#define OFF_CH  1024u                    // c_h     : 4
#define OFF_S0  1028u                    // s0_h    : 4
#define OFF_U   1056u                    // U[p][h][e] : 256*4*256
#define OFF_CTX (OFF_U   + 262144u)      // 256*256
#define OFF_MHA (OFF_CTX + 65536u)       // 256*256
#define OFF_LN  (OFF_MHA + 65536u)       // 256*256
#define OFF_H1  (OFF_LN  + 65536u)       // 256*1024
#define OFF_F2  (OFF_H1  + 262144u)      // 256*256

static __device__ __forceinline__ v8f wmma4(v2f a, v2f b, v8f c) {
  // D = A(16x4,f32) * B(4x16,f32) + C(16x16,f32)
  return __builtin_amdgcn_wmma_f32_16x16x4_f32(false, a, false, b, (short)0, c,
                                               false, false);
}

// ---------------------------------------------------------------------------
// k_prep: q0 = w_q@cls + b_q (cls row is identical for all pairs!),
//         R[h] = W_k_h^T q0_h, c_h = q0_h . b_k_h, s0_h = scale*(R.cls + c_h)
// ---------------------------------------------------------------------------
__global__ void __launch_bounds__(256) k_prep(const float* __restrict__ cls,
                                              const float* __restrict__ w_qkv,
                                              const float* __restrict__ b_qkv,
                                              float* __restrict__ ws) {
  __shared__ float clsL[256], q0L[256], RLs[1024];
  const int t = threadIdx.x;
  clsL[t] = cls[t];
  __syncthreads();
  float s = b_qkv[t];
  for (int e = 0; e < 256; ++e) s += w_qkv[(size_t)t * 256 + e] * clsL[e];
  q0L[t] = s;
  __syncthreads();
  for (int h = 0; h < 4; ++h) {
    float r = 0.f;
    for (int d = 0; d < 64; ++d)
      r += q0L[h * 64 + d] * w_qkv[(size_t)(256 + h * 64 + d) * 256 + t];
    RLs[h * 256 + t] = r;
    ws[OFF_R + h * 256 + t] = r;
  }
  __syncthreads();
  if (t < 4) {
    float c = 0.f;
    for (int d = 0; d < 64; ++d) c += q0L[t * 64 + d] * b_qkv[256 + t * 64 + d];
    float s0 = 0.f;
    for (int e = 0; e < 256; ++e) s0 += RLs[t * 256 + e] * clsL[e];
    ws[OFF_CH + t] = c;
    ws[OFF_S0 + t] = 0.125f * (s0 + c);  // scale = 1/sqrt(64)
  }
}

// ---------------------------------------------------------------------------
// k_attn: one block per (b,c). Online softmax over 32-key chunks; x read once.
// u = attn^T @ X computed with v_wmma_f32_16x16x4_f32 (A rows = 4 heads).
// ---------------------------------------------------------------------------
__global__ void __launch_bounds__(256) k_attn(const float* __restrict__ x,
                                              const int* __restrict__ real_cols,
                                              const int* __restrict__ real_rows,
                                              const float* __restrict__ cls,
                                              float* __restrict__ ws) {
  const int p = blockIdx.x;
  const int b = p >> 5, c = p & 31;
  const int tid = threadIdx.x;
  float* U = ws + OFF_U + (size_t)p * 1024;
  if (c >= real_cols[b]) {           // masked column: deterministic zeros
    for (int j = 0; j < 4; ++j) U[j * 256 + tid] = 0.f;
    return;
  }
  const int lane = tid & 31, wave = tid >> 5;

  __shared__ float xs[32][272];      // X chunk, padded pitch (bank-safe)
  __shared__ float RL[4][256];
  __shared__ float clsL[256];
  __shared__ float Apack[16][36];    // weight fragments, rows 4..15 stay 0
  __shared__ float part[32][8][4];
  __shared__ float sc[32][4];
  __shared__ float uacc[4][256];
  __shared__ float mmL[4], ddL[4], facL[4], s0L[4], chL[4];

  for (int h = 0; h < 4; ++h) RL[h][tid] = ws[OFF_R + h * 256 + tid];
  clsL[tid] = cls[tid];
  if (tid < 16)
    for (int k = 0; k < 36; ++k) Apack[tid][k] = 0.f;
  if (tid < 4) {
    chL[tid] = ws[OFF_CH + tid];
    s0L[tid] = ws[OFF_S0 + tid];
    mmL[tid] = s0L[tid];             // CLS key is always valid
    ddL[tid] = 1.f;
  }
  v8f acc0 = {0, 0, 0, 0, 0, 0, 0, 0};
  v8f acc1 = {0, 0, 0, 0, 0, 0, 0, 0};
  const int rr = real_rows[b];
  __syncthreads();

  const float* xp = x + (size_t)p * RR_ * E_;
  const int nch = (rr + 31) >> 5;
  for (int ch = 0; ch < nch; ++ch) {
    const int k0 = 1 + ch * 32;
    const int nval = min(32, rr - ch * 32);
    for (int i = 0; i < 32; ++i)     // coalesced stage (zeros past tail)
      xs[i][tid] = (i < nval) ? xp[(size_t)(k0 - 1 + i) * E_ + tid] : 0.f;
    __syncthreads();
    {  // scores: 8 threads per row, 4 heads each
      const int i = tid >> 3, s8 = tid & 7, e0 = s8 * 32;
      float p0 = 0, p1 = 0, p2 = 0, p3 = 0;
      for (int e = 0; e < 32; ++e) {
        const float xv = xs[i][e0 + e];
        p0 += RL[0][e0 + e] * xv; p1 += RL[1][e0 + e] * xv;
        p2 += RL[2][e0 + e] * xv; p3 += RL[3][e0 + e] * xv;
      }
      part[i][s8][0] = p0; part[i][s8][1] = p1;
      part[i][s8][2] = p2; part[i][s8][3] = p3;
    }
    __syncthreads();
    if (tid < 128) {
      const int i = tid >> 2, h = tid & 3;
      float s = 0.f;
      for (int j = 0; j < 8; ++j) s += part[i][j][h];
      sc[i][h] = 0.125f * (s + chL[h]);
    }
    __syncthreads();
    if (tid < 4) {                   // online max/denom per head
      const int h = tid;
      float m = mmL[h], mn = m;
      for (int i = 0; i < nval; ++i) mn = fmaxf(mn, sc[i][h]);
      const float f = __expf(m - mn);
      float d = ddL[h] * f;
      for (int i = 0; i < nval; ++i) d += __expf(sc[i][h] - mn);
      mmL[h] = mn; ddL[h] = d; facL[h] = f;
    }
    __syncthreads();
    if (tid < 128) {
      const int i = tid >> 2, h = tid & 3;
      Apack[h][i] = (i < nval) ? __expf(sc[i][h] - mmL[h]) : 0.f;
    }
    __syncthreads();
    {  // u += W_chunk @ X_chunk via WMMA; rescale accumulators first
      const int m = lane & 15;
      const int kb2 = (lane >> 4) << 1;
      const float f0 = facL[0], f1 = facL[1], f2v = facL[2], f3 = facL[3];
      acc0[0] *= f0; acc0[1] *= f1; acc0[2] *= f2v; acc0[3] *= f3;
      acc0[4] *= f0; acc0[5] *= f0; acc0[6] *= f0; acc0[7] *= f0;  // zero rows
      acc1[0] *= f0; acc1[1] *= f1; acc1[2] *= f2v; acc1[3] *= f3;
      acc1[4] *= f0; acc1[5] *= f0; acc1[6] *= f0; acc1[7] *= f0;
      const int n0 = wave * 32, n1 = n0 + 16;
      for (int kk = 0; kk < 8; ++kk) {
        const int kb = 4 * kk + kb2;
        v2f a;  a.x = Apack[m][kb];      a.y = Apack[m][kb + 1];
        v2f b0; b0.x = xs[kb][n0 + m];   b0.y = xs[kb + 1][n0 + m];
        v2f b1; b1.x = xs[kb][n1 + m];   b1.y = xs[kb + 1][n1 + m];
        acc0 = wmma4(a, b0, acc0);
        acc1 = wmma4(a, b1, acc1);
      }
    }
    __syncthreads();
  }
  if (lane < 16) {                   // rows 0..3 of D are the 4 heads
    const int m = lane & 15, n0 = wave * 32;
    for (int r = 0; r < 4; ++r) {
      uacc[r][n0 + m] = acc0[r];
      uacc[r][n0 + 16 + m] = acc1[r];
    }
  }
  __syncthreads();
  for (int h = 0; h < 4; ++h) {      // add CLS key, normalize
    const float w0 = __expf(s0L[h] - mmL[h]);
    U[h * 256 + tid] = (uacc[h][tid] + w0 * clsL[tid]) / ddL[h];
  }
}

// ---------------------------------------------------------------------------
// k_gemm: D[256][N] = op(A[256][K] @ Bw[N][K]^T + bias).  16 rows x 128 cols
// per block, one 16x16 tile per wave, K chunked by 64 through padded LDS.
// ---------------------------------------------------------------------------
__global__ void __launch_bounds__(256) k_gemm(const float* __restrict__ A, int lda,
                                              const float* __restrict__ Bw, int ldb,
                                              const float* __restrict__ bias,
                                              float* __restrict__ D, int ldd,
                                              int N, int K, int do_relu) {
  __shared__ float As[16][68];
  __shared__ float Bs[8][16][68];
  const int tid = threadIdx.x, lane = tid & 31, wave = tid >> 5;
  const int m0 = blockIdx.x * 16;
  const int n0 = blockIdx.y * 128 + wave * 16;
  const bool active = (n0 < N);
  v8f acc = {0, 0, 0, 0, 0, 0, 0, 0};
  for (int kc = 0; kc < K; kc += 64) {
    for (int j = 0; j < 4; ++j) {
      const int idx = tid + j * 256, r = idx >> 6, cc = idx & 63;
      As[r][cc] = A[(size_t)(m0 + r) * lda + kc + cc];
    }
    if (active) {
      for (int j = 0; j < 32; ++j) {
        const int idx = lane + j * 32, r = idx >> 6, cc = idx & 63;
        Bs[wave][r][cc] = Bw[(size_t)(n0 + r) * ldb + kc + cc];
      }
    }
    __syncthreads();
    if (active) {
      const int m = lane & 15;
      const int kb2 = (lane >> 4) << 1;
      for (int kk = 0; kk < 16; ++kk) {
        const int kb = 4 * kk + kb2;
        v2f a;  a.x = As[m][kb];        a.y = As[m][kb + 1];
        v2f bf; bf.x = Bs[wave][m][kb]; bf.y = Bs[wave][m][kb + 1];
        acc = wmma4(a, bf, acc);
      }
    }
    __syncthreads();
  }
  if (active) {
    const int m = lane & 15, hi = lane >> 4;
    const float bv = bias ? bias[n0 + m] : 0.f;
    for (int r = 0; r < 8; ++r) {
      float v = acc[r] + bv;
      if (do_relu) v = fmaxf(v, 0.f);
      D[(size_t)(m0 + r + (hi << 3)) * ldd + n0 + m] = v;
    }
  }
}

// ---------------------------------------------------------------------------
__global__ void __launch_bounds__(256) k_ln(const float* __restrict__ mha,
                                            const float* __restrict__ g,
                                            const float* __restrict__ bb,
                                            float* __restrict__ ln) {
  __shared__ float red[256];
  __shared__ float mu_s, var_s;
  const int p = blockIdx.x, e = threadIdx.x;
  const float v = mha[p * 256 + e];
  red[e] = v;
  __syncthreads();
  for (int s = 128; s > 0; s >>= 1) { if (e < s) red[e] += red[e + s]; __syncthreads(); }
  if (e == 0) mu_s = red[0] * (1.f / 256.f);
  __syncthreads();
  const float d = v - mu_s;
  red[e] = d * d;
  __syncthreads();
  for (int s = 128; s > 0; s >>= 1) { if (e < s) red[e] += red[e + s]; __syncthreads(); }
  if (e == 0) var_s = red[0] * (1.f / 256.f);
  __syncthreads();
  ln[p * 256 + e] = d * rsqrtf(var_s + 1e-5f) * g[e] + bb[e];
}

__global__ void __launch_bounds__(256) k_final(const float* __restrict__ ln,
                                               const float* __restrict__ f2,
                                               const int* __restrict__ real_cols,
                                               float* __restrict__ out) {
  const int p = blockIdx.x, e = threadIdx.x;
  const int b = p >> 5, c = p & 31;
  const float v = ln[p * 256 + e] + f2[p * 256 + e];
  out[p * 256 + e] = (c < real_cols[b]) ? v : 0.f;
}

// ---------------------------------------------------------------------------
extern "C" void kernel_launch(void* const* d_in, const int* in_sizes, int n_in,
                              void* d_out, int out_size, void* d_ws, size_t ws_size,
                              hipStream_t stream) {
  const float* x   = (const float*)d_in[0];
  const int* rcol  = (const int*)d_in[1];
  const int* rrow  = (const int*)d_in[2];
  const float* cls = (const float*)d_in[3];
  const float* wqkv= (const float*)d_in[4];
  const float* bqkv= (const float*)d_in[5];
  const float* wo  = (const float*)d_in[6];
  const float* bo  = (const float*)d_in[7];
  const float* lng = (const float*)d_in[8];
  const float* lnb = (const float*)d_in[9];
  const float* w1  = (const float*)d_in[10];
  const float* b1  = (const float*)d_in[11];
  const float* w2  = (const float*)d_in[12];
  const float* b2  = (const float*)d_in[13];
  float* ws  = (float*)d_ws;
  float* out = (float*)d_out;

  k_prep<<<1, 256, 0, stream>>>(cls, wqkv, bqkv, ws);
  k_attn<<<256, 256, 0, stream>>>(x, rcol, rrow, cls, ws);
  // ctx = U_h @ W_v_h^T + b_v  (4 head GEMMs, N=64, K=256)
  for (int h = 0; h < 4; ++h) {
    k_gemm<<<dim3(16, 1), 256, 0, stream>>>(
        ws + OFF_U + h * 256, 1024,
        wqkv + (size_t)(512 + h * 64) * 256, 256,
        bqkv + 512 + h * 64,
        ws + OFF_CTX + h * 64, 256, 64, 256, 0);
  }
  // mha = ctx @ w_out^T + b_out
  k_gemm<<<dim3(16, 2), 256, 0, stream>>>(ws + OFF_CTX, 256, wo, 256, bo,
                                          ws + OFF_MHA, 256, 256, 256, 0);
  k_ln<<<256, 256, 0, stream>>>(ws + OFF_MHA, lng, lnb, ws + OFF_LN);
  // h1 = relu(ln @ w1^T + b1)
  k_gemm<<<dim3(16, 8), 256, 0, stream>>>(ws + OFF_LN, 256, w1, 256, b1,
                                          ws + OFF_H1, 1024, 1024, 256, 1);
  // f2 = h1 @ w2^T + b2
  k_gemm<<<dim3(16, 2), 256, 0, stream>>>(ws + OFF_H1, 1024, w2, 1024, b2,
                                          ws + OFF_F2, 256, 256, 1024, 0);
  k_final<<<256, 256, 0, stream>>>(ws + OFF_LN, ws + OFF_F2, rcol, out);
}